// GE2E_Loss_34900904247398
// MI455X (gfx1250) — compile-verified
//
#include <hip/hip_runtime.h>
#include <math.h>

// ---------------------------------------------------------------------------
// GE2E softmax-contrast loss for MI455X (gfx1250, wave32, WMMA + TDM)
//   emb: [N,D] fp32, y: [N] int32 (C classes), w,b scalars
//   loss = sum_i [ logsumexp_col(y_i) - sim[i, y_i] ]  (softmax over batch axis)
// Main GEMM S = e @ cen^T uses V_WMMA_F32_16X16X4_F32; centroid tiles are
// staged into LDS by the Tensor Data Mover with hardware bank-padding.
// ---------------------------------------------------------------------------

typedef __attribute__((ext_vector_type(2))) float v2f;
typedef __attribute__((ext_vector_type(8))) float v8f;
typedef __attribute__((ext_vector_type(4))) unsigned int v4u;
typedef __attribute__((ext_vector_type(8))) int v8i_t;
typedef __attribute__((ext_vector_type(4))) int v4i_t;

namespace {
constexpr int kN = 16384;   // batch
constexpr int kD = 512;     // embedding dim
constexpr int kC = 256;     // classes
constexpr int kChunks = 8;  // deterministic partial-sum chunks for segment sum
constexpr float kEpsNorm = 1e-12f;
constexpr float kEpsCos  = 1e-8f;
// LDS pitch for B tile: 512 data + 2 pad per 256 DWORDs -> 516 floats/row.
// 516 % 64 == 4 => 16 tile rows hit 16 distinct LDS banks on fragment reads,
// and the pad offset (+2 for d>=256) preserves 8-byte alignment of b64 reads.
constexpr int BPAD = kD + 4;
__device__ __forceinline__ int padIdx(int d) { return d + ((d >> 8) << 1); }
}

// --- Kernel 1: row L2 normalize; one wave per row (32 lanes x 16 floats) ----
__global__ void k_normalize(const float* __restrict__ emb,
                            float* __restrict__ e,
                            float* __restrict__ enorm) {
    const int row  = blockIdx.x * 8 + (threadIdx.x >> 5);
    const int lane = threadIdx.x & 31;
    const float4* src = (const float4*)(emb + (size_t)row * kD);
    float4 v[4];
    float ss = 0.0f;
#pragma unroll
    for (int j = 0; j < 4; ++j) {
        v[j] = src[lane + 32 * j];
        ss += v[j].x * v[j].x + v[j].y * v[j].y + v[j].z * v[j].z + v[j].w * v[j].w;
    }
#pragma unroll
    for (int off = 16; off > 0; off >>= 1) ss += __shfl_down(ss, off, 32);
    ss = __shfl(ss, 0, 32);
    const float nrm   = sqrtf(ss);
    const float scale = 1.0f / fmaxf(nrm, kEpsNorm);
    float4* dst = (float4*)(e + (size_t)row * kD);
#pragma unroll
    for (int j = 0; j < 4; ++j) {
        dst[lane + 32 * j] = make_float4(v[j].x * scale, v[j].y * scale,
                                         v[j].z * scale, v[j].w * scale);
    }
    if (lane == 0) enorm[row] = nrm * scale;   // == ||e_row|| (~1.0)
}

// --- Kernel 2a: deterministic per-class partial sums over batch chunks ------
__global__ void k_csum_part(const float* __restrict__ e,
                            const int* __restrict__ y,
                            float* __restrict__ part,
                            float* __restrict__ partcnt) {
    const int c = blockIdx.x;
    const int j = blockIdx.y;
    const int t = threadIdx.x;           // 256 threads cover d and d+256
    const int i0 = j * (kN / kChunks);
    const int i1 = i0 + (kN / kChunks);
    float a0 = 0.0f, a1 = 0.0f;
    int count = 0;
    for (int i = i0; i < i1; ++i) {
        if (y[i] == c) {
            const float* er = e + (size_t)i * kD;
            a0 += er[t];
            a1 += er[t + 256];
            ++count;
        }
    }
    const size_t pb = ((size_t)c * kChunks + j) * kD;
    part[pb + t]       = a0;
    part[pb + t + 256] = a1;
    if (t == 0) partcnt[c * kChunks + j] = (float)count;
}

// --- Kernel 2b: reduce chunks -> cen (=csum/cnt), count, ||csum||^2, ||cen|| -
__global__ void k_csum_reduce(const float* __restrict__ part,
                              const float* __restrict__ partcnt,
                              float* __restrict__ cen,
                              float* __restrict__ cntf,
                              float* __restrict__ csq,
                              float* __restrict__ cnorm) {
    const int c = blockIdx.x;
    const int t = threadIdx.x;
    float a0 = 0.0f, a1 = 0.0f;
#pragma unroll
    for (int j = 0; j < kChunks; ++j) {
        const size_t pb = ((size_t)c * kChunks + j) * kD;
        a0 += part[pb + t];
        a1 += part[pb + t + 256];
    }
    float fc = 0.0f;                     // uniform across threads
#pragma unroll
    for (int j = 0; j < kChunks; ++j) fc += partcnt[c * kChunks + j];
    const float inv = 1.0f / fc;
    cen[(size_t)c * kD + t]       = a0 * inv;
    cen[(size_t)c * kD + t + 256] = a1 * inv;
    __shared__ float red[256];
    red[t] = a0 * a0 + a1 * a1;          // ||csum||^2 pieces
    __syncthreads();
    for (int s = 128; s > 0; s >>= 1) {
        if (t < s) red[t] += red[t + s];
        __syncthreads();
    }
    if (t == 0) {
        const float sq = red[0];
        cntf[c]  = fc;
        csq[c]   = sq;
        cnorm[c] = sqrtf(sq) * inv;      // ||csum/cnt||
    }
}

// --- Kernel 3: WMMA GEMM  S = e @ cen^T  + similarity epilogue ---------------
// Block: 256 threads = 8 waves; block tile = 128 rows x 16 classes.
// Wave tile = 16x16, K-loop over D in steps of 4 with V_WMMA_F32_16X16X4_F32.
// B tile (16 centroids) staged into LDS by TDM with hardware pad for banks.
// Output stored TRANSPOSED: sim_t[class][row] for coalesced column reductions.
__global__ void k_gemm_sim(const float* __restrict__ e,
                           const int* __restrict__ y,
                           const float* __restrict__ cen,
                           const float* __restrict__ cntf,
                           const float* __restrict__ csq,
                           const float* __restrict__ cnorm,
                           const float* __restrict__ enorm,
                           const float* __restrict__ wp,
                           const float* __restrict__ bp,
                           float* __restrict__ sim_t) {
    __shared__ float Bs[16 * BPAD];

    const int c0   = blockIdx.x * 16;
    const int i0   = blockIdx.y * 128;
    const int tid  = threadIdx.x;
    const int wave = tid >> 5;
    const int lane = tid & 31;

#if __has_builtin(__builtin_amdgcn_tensor_load_to_lds)
    {
        // Each wave DMA-loads 2 centroid rows (2 x 512 f32) into padded LDS.
        // D# group1: data_size=4B(2), pad_enable=1, pad_interval=256dw(7),
        // pad_amount=2dw(1); tile 512x2; tensor_dim0(_stride)=512, dim1=2.
        const float* gsrc = cen + (size_t)(c0 + wave * 2) * kD;
        const unsigned long long ga = (unsigned long long)(uintptr_t)gsrc;
        const unsigned int ldsOff =
            (unsigned int)(uintptr_t)(&Bs[0]) + (unsigned int)(wave * 2 * BPAD * 4);
        const v4u g0 = { 1u,                                  // count=1 (valid D#)
                         ldsOff,                              // lds_addr
                         (unsigned int)(ga & 0xFFFFFFFFu),    // global_addr[31:0]
                         (unsigned int)((ga >> 32) & 0x01FFFFFFu) | (2u << 30) }; // addr[56:32] | type=2
        const v8i_t g1 = { (int)0x03D20000,   // data_size=2, pad_en, intv=7, amt=1
                           (int)(512 << 16),  // tensor_dim0[15:0]=512
                           (int)(2 << 16),    // tensor_dim1[15:0]=2
                           (int)(512 << 16),  // tile_dim0=512
                           2,                 // tile_dim1=2
                           512,               // tensor_dim0_stride[31:0]=512
                           0, 0 };
        const v4i_t gz4 = { 0, 0, 0, 0 };
        const v8i_t gz8 = { 0, 0, 0, 0, 0, 0, 0, 0 };
        // 6-arg toolchain form: (g0, g1, g2, g3, g4, cpol)
        __builtin_amdgcn_tensor_load_to_lds(g0, g1, gz4, gz4, gz8, 0);
        __builtin_amdgcn_s_wait_tensorcnt(0);
    }
#else
    for (int idx = tid; idx < 16 * kD; idx += 256) {
        const int cc = idx >> 9;
        const int d  = idx & (kD - 1);
        Bs[cc * BPAD + padIdx(d)] = cen[(size_t)(c0 + cc) * kD + d];
    }
#endif
    __syncthreads();

    const int r16   = lane & 15;                 // A-row / B-col within tile
    const int khalf = (lane >> 4) * 2;           // K sub-offset per lane half
    const int iBase = i0 + wave * 16;

    const float* arow  = e  + (size_t)(iBase + r16) * kD;
    const float* bsrow = Bs + r16 * BPAD;

    v8f acc = {};
    for (int k0 = 0; k0 < kD; k0 += 64) {        // 16 WMMAs per chunk
        if (k0 + 64 < kD)
            __builtin_prefetch(arow + k0 + 64, 0, 3);   // global_prefetch_b8
#pragma unroll
        for (int k = 0; k < 64; k += 4) {
            const int kk = k0 + k + khalf;
            v2f a = *(const v2f*)(arow + kk);
            v2f b = *(const v2f*)(bsrow + padIdx(kk));
            // D(16x16,f32) = A(16x4,f32) * B(4x16,f32) + C
            acc = __builtin_amdgcn_wmma_f32_16x16x4_f32(
                /*neg_a=*/false, a, /*neg_b=*/false, b,
                /*c_mod=*/(short)0, acc, /*reuse_a=*/false, /*reuse_b=*/false);
        }
    }

    // Epilogue: cosine sims (+ leave-one-out self sim), scale by w, add b.
    const float w     = wp[0];
    const float b     = bp[0];
    const int   cls   = c0 + r16;                // D-matrix col = lane & 15
    const float cn    = cnorm[cls];
    const float fcnt  = cntf[cls];
    const float sq    = csq[cls];
    const float invm1 = 1.0f / (fcnt - 1.0f);
    const int   iRow  = iBase + ((lane >> 4) << 3);   // 8 consecutive rows/lane

    const int4   ya = *(const int4*)(y + iRow);
    const int4   yb = *(const int4*)(y + iRow + 4);
    const float4 ea = *(const float4*)(enorm + iRow);
    const float4 eb = *(const float4*)(enorm + iRow + 4);
    const int   yv[8] = { ya.x, ya.y, ya.z, ya.w, yb.x, yb.y, yb.z, yb.w };
    const float ev[8] = { ea.x, ea.y, ea.z, ea.w, eb.x, eb.y, eb.z, eb.w };

    float ov[8];
#pragma unroll
    for (int v = 0; v < 8; ++v) {                // acc[v] is row iRow+v, col cls
        const float raw = acc[v];                // e_i . cen_cls
        const float en  = ev[v];
        float val;
        if (yv[v] == cls) {
            // cen_self = (csum - e_i)/(cnt-1); everything from scalars:
            const float dcs = raw * fcnt;        // e_i . csum
            const float en2 = en * en;
            const float num = (dcs - en2) * invm1;
            const float sn  = sqrtf(fmaxf(sq - 2.0f * dcs + en2, 0.0f)) * invm1;
            val = num / fmaxf(en * sn, kEpsCos);
        } else {
            val = raw / fmaxf(en * cn, kEpsCos);
        }
        ov[v] = val * w + b;
    }
    float4* dst = (float4*)(sim_t + (size_t)cls * kN + iRow);
    dst[0] = make_float4(ov[0], ov[1], ov[2], ov[3]);
    dst[1] = make_float4(ov[4], ov[5], ov[6], ov[7]);
}

// --- Kernel 4: per-class column logsumexp + self-entry sum -------------------
__global__ void k_colreduce(const float* __restrict__ sim_t,
                            const int* __restrict__ y,
                            const float* __restrict__ cntf,
                            float* __restrict__ partial) {
    const int c = blockIdx.x;
    const int t = threadIdx.x;
    const float* col = sim_t + (size_t)c * kN;
    __shared__ float red[256];

    float mx = -__builtin_inff();
    for (int i = t; i < kN; i += 256) mx = fmaxf(mx, col[i]);
    red[t] = mx;
    __syncthreads();
    for (int s = 128; s > 0; s >>= 1) {
        if (t < s) red[t] = fmaxf(red[t], red[t + s]);
        __syncthreads();
    }
    mx = red[0];
    __syncthreads();

    float se = 0.0f, selfs = 0.0f;
    for (int i = t; i < kN; i += 256) {
        const float x = col[i];
        se += expf(x - mx);
        if (y[i] == c) selfs += x;
    }
    red[t] = se;
    __syncthreads();
    for (int s = 128; s > 0; s >>= 1) {
        if (t < s) red[t] += red[t + s];
        __syncthreads();
    }
    const float seTot = red[0];
    __syncthreads();
    red[t] = selfs;
    __syncthreads();
    for (int s = 128; s > 0; s >>= 1) {
        if (t < s) red[t] += red[t + s];
        __syncthreads();
    }
    if (t == 0) {
        const float lse = mx + logf(seTot);
        partial[c] = cntf[c] * lse - red[0];
    }
}

// --- Kernel 5: final sum over classes ---------------------------------------
__global__ void k_final(const float* __restrict__ partial, float* __restrict__ out) {
    const int t = threadIdx.x;                   // 256 == kC
    __shared__ float red[256];
    red[t] = partial[t];
    __syncthreads();
    for (int s = 128; s > 0; s >>= 1) {
        if (t < s) red[t] += red[t + s];
        __syncthreads();
    }
    if (t == 0) out[0] = red[0];
}

// ---------------------------------------------------------------------------
extern "C" void kernel_launch(void* const* d_in, const int* in_sizes, int n_in,
                              void* d_out, int out_size, void* d_ws, size_t ws_size,
                              hipStream_t stream) {
    const float* emb = (const float*)d_in[0];
    const int*   y   = (const int*)d_in[1];
    const float* wp  = (const float*)d_in[2];
    const float* bp  = (const float*)d_in[3];
    float* out = (float*)d_out;

    // Workspace carve (~55 MB)
    float* p       = (float*)d_ws;
    float* e       = p;  p += (size_t)kN * kD;          // normalized embeddings
    float* enorm   = p;  p += kN;                       // ||e_i||
    float* part    = p;  p += (size_t)kC * kChunks * kD;// partial centroid sums
    float* partcnt = p;  p += (size_t)kC * kChunks;     // partial counts
    float* cen     = p;  p += (size_t)kC * kD;          // centroids (csum/cnt)
    float* cntf    = p;  p += kC;                       // counts (float)
    float* csq     = p;  p += kC;                       // ||csum||^2
    float* cnorm   = p;  p += kC;                       // ||centroid||
    float* simt    = p;  p += (size_t)kC * kN;          // sim transposed [C][N]
    float* partial = p;  p += kC;                       // per-class loss parts

    k_normalize  <<<kN / 8, 256, 0, stream>>>(emb, e, enorm);
    k_csum_part  <<<dim3(kC, kChunks), 256, 0, stream>>>(e, y, part, partcnt);
    k_csum_reduce<<<kC, 256, 0, stream>>>(part, partcnt, cen, cntf, csq, cnorm);
    k_gemm_sim   <<<dim3(kC / 16, kN / 128), 256, 0, stream>>>(
        e, y, cen, cntf, csq, cnorm, enorm, wp, bp, simt);
    k_colreduce  <<<kC, 256, 0, stream>>>(simt, y, cntf, partial);
    k_final      <<<1, 256, 0, stream>>>(partial, out);
}